// CandidateScorer_87780541596105
// MI455X (gfx1250) — compile-verified
//
#include <hip/hip_runtime.h>
#include <math.h>

// ---------------------------------------------------------------------------
// Problem constants (fixed by the reference)
// ---------------------------------------------------------------------------
#define NNODES   50000
#define NEDGES   800000
#define NPAIRS   200000
#define NODE_IN  128
#define EDGE_IN  16
#define CAND_IN  32
#define HID      64
#define E2       (NEDGES + NNODES)   // edges + self loops
#define NEG_SLOPE 0.2f

typedef float v2f __attribute__((ext_vector_type(2)));
typedef float v8f __attribute__((ext_vector_type(8)));

// Monotonic float<->int mapping so we can use integer atomicMax for
// segment-max of (possibly negative) logits.
__device__ __forceinline__ int mono_f2i(float f) {
  int b = __float_as_int(f);
  return b >= 0 ? b : (b ^ 0x7FFFFFFF);
}
__device__ __forceinline__ float mono_i2f(int m) {
  int b = m >= 0 ? m : (m ^ 0x7FFFFFFF);
  return __int_as_float(b);
}

// ---------------------------------------------------------------------------
// Utility fills
// ---------------------------------------------------------------------------
__global__ void fill_i32(int* __restrict__ p, int n, int v) {
  int i = blockIdx.x * blockDim.x + threadIdx.x;
  if (i < n) p[i] = v;
}

// ---------------------------------------------------------------------------
// Stage 0: self-loop edge attr = segment_mean(edge_attr by dst)
// ---------------------------------------------------------------------------
__global__ void deg_scatter(const int* __restrict__ dst,
                            const float* __restrict__ eattr,
                            float* __restrict__ cnt,
                            float* __restrict__ loop_sum) {
  int t = blockIdx.x * blockDim.x + threadIdx.x;
  if (t >= NEDGES * EDGE_IN) return;
  int e = t >> 4, k = t & 15;
  int d = dst[e];
  if (k == 0) atomicAdd(&cnt[d], 1.0f);
  atomicAdd(&loop_sum[d * EDGE_IN + k], eattr[t]);
}

__global__ void loop_div(float* __restrict__ loop_attr, const float* __restrict__ cnt) {
  int t = blockIdx.x * blockDim.x + threadIdx.x;
  if (t >= NNODES * EDGE_IN) return;
  loop_attr[t] /= fmaxf(cnt[t >> 4], 1.0f);
}

// ---------------------------------------------------------------------------
// fp32 WMMA GEMM, one wave per 16-row strip with TN column tiles:
//   C[M, TN*16] = act(A[M,K] @ W[K, TN*16] + bias)
// The A fragment is loaded ONCE per K-step and reused across all TN B tiles
// (TN independent accumulators -> TN back-to-back v_wmma, hides RAW latency).
// Requires M%16==0, K%4==0.
// fp32 frag layout (ISA 7.12.2):
//   A 16x4 : lane(l<16): row=l, K={0,1}; lane(l>=16): row=l-16, K={2,3}
//   B 4x16 : lane(l<16): col=l, K={0,1}; lane(l>=16): col=l-16, K={2,3}
//   C/D    : vgpr i: rows i (lanes 0-15) / i+8 (lanes 16-31), col=lane&15
// ---------------------------------------------------------------------------
template <int TN>
__global__ void wmma_gemm_rows(const float* __restrict__ A,
                               const float* __restrict__ W,
                               const float* __restrict__ bias,
                               float* __restrict__ C,
                               int M, int K, int relu) {
  const int Nout = TN * 16;
  int wid  = (blockIdx.x * blockDim.x + threadIdx.x) >> 5;   // strip index
  int lane = threadIdx.x & 31;
  if (wid >= (M >> 4)) return;                 // wave-uniform guard
  int half = lane >> 4, r = lane & 15;
  const float* arow = A + (size_t)(wid * 16 + r) * K;

  v8f acc[TN];
  #pragma unroll
  for (int t = 0; t < TN; ++t) acc[t] = v8f{0.f,0.f,0.f,0.f,0.f,0.f,0.f,0.f};

  #pragma unroll 2
  for (int k0 = 0; k0 < K; k0 += 4) {
    int ka = k0 + half * 2;
    v2f a;
    a.x = arow[ka];
    a.y = arow[ka + 1];
    const float* w0 = W + (size_t)ka * Nout + r;
    #pragma unroll
    for (int t = 0; t < TN; ++t) {
      v2f b;
      b.x = w0[t * 16];
      b.y = w0[t * 16 + Nout];
      acc[t] = __builtin_amdgcn_wmma_f32_16x16x4_f32(false, a, false, b,
                                                     (short)0, acc[t], false, false);
    }
  }
  #pragma unroll
  for (int t = 0; t < TN; ++t) {
    int colC = t * 16 + r;
    float bv = bias ? bias[colC] : 0.0f;
    #pragma unroll
    for (int i = 0; i < 8; ++i) {
      int row = wid * 16 + i + half * 8;
      float v = acc[t][i] + bv;
      if (relu) v = fmaxf(v, 0.0f);
      C[(size_t)row * Nout + colC] = v;
    }
  }
}

// ---------------------------------------------------------------------------
// GATv2 attention logits. One wave per (edge or self-loop).
// logit[e,h] = sum_c leaky( xl[src,h,c] + xr[dst,h,c] + (ea@We)[h,c] ) * att[h,c]
// Also segment-max into lmax via monotonic-int atomicMax.
// ---------------------------------------------------------------------------
template <int HEADS, int HD>
__global__ void gat_logits(const int* __restrict__ src, const int* __restrict__ dst,
                           const float* __restrict__ eattr, const float* __restrict__ loop_attr,
                           const float* __restrict__ xl, const float* __restrict__ xr,
                           const float* __restrict__ We,  const float* __restrict__ att,
                           float* __restrict__ logits, int* __restrict__ lmax) {
  int wid  = (blockIdx.x * blockDim.x + threadIdx.x) >> 5;
  int lane = threadIdx.x & 31;
  if (wid >= E2) return;                       // wave-uniform
  int s, d; const float* ea;
  if (wid < NEDGES) { s = src[wid]; d = dst[wid]; ea = eattr + (size_t)wid * EDGE_IN; }
  else              { s = wid - NEDGES; d = s;    ea = loop_attr + (size_t)s * EDGE_IN; }

  float eav[EDGE_IN];
  #pragma unroll
  for (int t = 0; t < EDGE_IN; ++t) eav[t] = ea[t];

  const int CPT = HD / 32;                     // features per lane
  const float* xls = xl + (size_t)s * HD;
  const float* xrd = xr + (size_t)d * HD;
  float part = 0.f;
  #pragma unroll
  for (int jj = 0; jj < CPT; ++jj) {
    int j = lane * CPT + jj;
    float ee = 0.f;
    #pragma unroll
    for (int t = 0; t < EDGE_IN; ++t) ee += eav[t] * We[t * HD + j];
    float m = xls[j] + xrd[j] + ee;
    m = (m > 0.f) ? m : NEG_SLOPE * m;
    part += m * att[j];
  }
  const int GROUP = 32 / HEADS;                // lanes per head
  #pragma unroll
  for (int mask = 1; mask < GROUP; mask <<= 1)
    part += __shfl_xor(part, mask, 32);
  if ((lane & (GROUP - 1)) == 0) {
    int h = lane / GROUP;
    logits[(size_t)wid * HEADS + h] = part;
    atomicMax(&lmax[d * HEADS + h], mono_f2i(part));
  }
}

// exp(logit - lmax[dst]) in place, accumulate softmax denominator.
template <int HEADS>
__global__ void gat_expsum(const int* __restrict__ dst,
                           float* __restrict__ logits,
                           const int* __restrict__ lmax,
                           float* __restrict__ denom) {
  int t = blockIdx.x * blockDim.x + threadIdx.x;
  if (t >= E2 * HEADS) return;
  int e2 = t / HEADS, h = t % HEADS;
  int d = (e2 < NEDGES) ? dst[e2] : (e2 - NEDGES);
  float ex = expf(logits[t] - mono_i2f(lmax[d * HEADS + h]));
  logits[t] = ex;
  atomicAdd(&denom[d * HEADS + h], ex);
}

// out[dst, j] += (ex/denom) * xl[src, j]
template <int HEADS, int HD>
__global__ void gat_agg(const int* __restrict__ src, const int* __restrict__ dst,
                        const float* __restrict__ logits, const float* __restrict__ denom,
                        const float* __restrict__ xl, float* __restrict__ out) {
  size_t t = (size_t)blockIdx.x * blockDim.x + threadIdx.x;
  if (t >= (size_t)E2 * HD) return;
  int j  = (int)(t % HD);
  int e2 = (int)(t / HD);
  int s, d;
  if (e2 < NEDGES) { s = src[e2]; d = dst[e2]; } else { s = d = e2 - NEDGES; }
  int h = j / (HD / HEADS);
  float alpha = logits[(size_t)e2 * HEADS + h] / denom[d * HEADS + h];
  atomicAdd(&out[(size_t)d * HD + j], alpha * xl[(size_t)s * HD + j]);
}

__global__ void bias_relu(float* __restrict__ x, const float* __restrict__ bias,
                          int total, int dim) {
  int t = blockIdx.x * blockDim.x + threadIdx.x;
  if (t >= total) return;
  x[t] = fmaxf(x[t] + bias[t % dim], 0.0f);
}

// ---------------------------------------------------------------------------
// Score hidden layer: s_hid = relu([hu | hv | |hu-hv| | hc] @ W1 + b1)
// K=256 WMMA GEMM; the A-fragment is synthesized on the fly from gathers
// (no z buffer) and reused across all 4 column tiles (one wave per 16 rows).
// The K loop is split into 4 compile-time segments (branch-free bodies);
// unroll is capped at 4 K-steps so the scheduler cannot hoist the whole K
// range of W loads and spill VGPRs to scratch.
// ---------------------------------------------------------------------------
__global__ void score_hidden(const int* __restrict__ pairs,
                             const float* __restrict__ h2,
                             const float* __restrict__ hc,
                             const float* __restrict__ W,    // 256 x 64
                             const float* __restrict__ bias, // 64
                             float* __restrict__ shid) {
  const int TN = HID / 16;                     // 4
  int wid  = (blockIdx.x * blockDim.x + threadIdx.x) >> 5;
  int lane = threadIdx.x & 31;
  if (wid >= (NPAIRS / 16)) return;            // wave-uniform
  int half = lane >> 4, r = lane & 15;
  int row = wid * 16 + r;
  int u = pairs[row * 2 + 0];
  int v = pairs[row * 2 + 1];
  const float* hu  = h2 + (size_t)u * HID;
  const float* hv  = h2 + (size_t)v * HID;
  const float* hcr = hc + (size_t)row * HID;

  v8f acc[TN];
  #pragma unroll
  for (int t = 0; t < TN; ++t) acc[t] = v8f{0.f,0.f,0.f,0.f,0.f,0.f,0.f,0.f};

  // one WMMA K-step: A values (ax, ay) at global K position kglob..kglob+1
  auto step = [&](float ax, float ay, int kglob) {
    v2f a; a.x = ax; a.y = ay;
    const float* w0 = W + (size_t)kglob * HID + r;
    #pragma unroll
    for (int t = 0; t < TN; ++t) {
      v2f b;
      b.x = w0[t * 16];
      b.y = w0[t * 16 + HID];
      acc[t] = __builtin_amdgcn_wmma_f32_16x16x4_f32(false, a, false, b,
                                                     (short)0, acc[t], false, false);
    }
  };

  // segment 0: hu
  #pragma unroll 4
  for (int k0 = 0; k0 < 64; k0 += 4) {
    int ka = k0 + half * 2;
    step(hu[ka], hu[ka + 1], ka);
  }
  // segment 1: hv
  #pragma unroll 4
  for (int k0 = 0; k0 < 64; k0 += 4) {
    int ka = k0 + half * 2;
    step(hv[ka], hv[ka + 1], 64 + ka);
  }
  // segment 2: |hu - hv|  (hu/hv lines already hot in WGP$)
  #pragma unroll 4
  for (int k0 = 0; k0 < 64; k0 += 4) {
    int ka = k0 + half * 2;
    step(fabsf(hu[ka] - hv[ka]), fabsf(hu[ka + 1] - hv[ka + 1]), 128 + ka);
  }
  // segment 3: hc
  #pragma unroll 4
  for (int k0 = 0; k0 < 64; k0 += 4) {
    int ka = k0 + half * 2;
    step(hcr[ka], hcr[ka + 1], 192 + ka);
  }

  #pragma unroll
  for (int t = 0; t < TN; ++t) {
    int colC = t * 16 + r;
    float bv = bias[colC];
    #pragma unroll
    for (int i = 0; i < 8; ++i) {
      int rr = wid * 16 + i + half * 8;
      shid[(size_t)rr * HID + colC] = fmaxf(acc[t][i] + bv, 0.0f);
    }
  }
}

// Final 64 -> 1 projection.
__global__ void score_final(const float* __restrict__ shid,
                            const float* __restrict__ W2,  // 64 x 1
                            const float* __restrict__ b2,
                            float* __restrict__ out) {
  int i = blockIdx.x * blockDim.x + threadIdx.x;
  if (i >= NPAIRS) return;
  const float* rowp = shid + (size_t)i * HID;
  float acc = b2[0];
  #pragma unroll
  for (int j = 0; j < HID; ++j) acc += rowp[j] * W2[j];
  out[i] = acc;
}

// ---------------------------------------------------------------------------
// Workspace layout (float offsets). Stage-3 overlays live on dead stage-1/2
// buffers. Peak = 32.5M floats = 130 MB.
// ---------------------------------------------------------------------------
static const size_t F_LOOP = 0;          //   800000  self-loop attrs (N x 16)
static const size_t F_CNT  = 800000;     //    50000  in-degree
static const size_t F_XL1  = 850000;     //  6400000  xl layer1 (N x 128)
static const size_t F_XR1  = 7250000;    //  6400000  xr layer1
static const size_t F_H1   = 13650000;   //  6400000  out1 accum -> h1
static const size_t F_LG1  = 20050000;   //  1700000  logits1 (E2 x 2)
static const size_t F_LM1  = 21750000;   //   100000  lmax1 (int)
static const size_t F_DN1  = 21850000;   //   100000  denom1
static const size_t F_XL2  = 21950000;   //  3200000  xl layer2 (N x 64)
static const size_t F_XR2  = 25150000;   //  3200000  xr layer2
static const size_t F_H2   = 28350000;   //  3200000  out2 accum -> h2
static const size_t F_LG2  = 31550000;   //   850000  logits2 (E2)
static const size_t F_LM2  = 32400000;   //    50000  lmax2 (int)
static const size_t F_DN2  = 32450000;   //    50000  denom2
// overlays (everything from stages 1/2 except h2 is dead by stage 3):
static const size_t F_HC1  = F_XL1;      // 12800000  (covers xl1+xr1)
static const size_t F_HC   = F_H1;       // 12800000  (covers h1..xr2 part)
static const size_t F_SHID = 0;          // 12800000  (covers loop..xr1 part)

static inline int cdiv(long a, long b) { return (int)((a + b - 1) / b); }

extern "C" void kernel_launch(void* const* d_in, const int* in_sizes, int n_in,
                              void* d_out, int out_size, void* d_ws, size_t ws_size,
                              hipStream_t stream) {
  (void)in_sizes; (void)n_in; (void)out_size; (void)ws_size;

  // Inputs in setup_inputs() dict order (params nested in insertion order).
  const float* x         = (const float*)d_in[0];   // N x 128
  const int*   edge_idx  = (const int*)  d_in[1];   // 2 x E
  const float* edge_attr = (const float*)d_in[2];   // E x 16
  const int*   pairs     = (const int*)  d_in[3];   // P x 2
  const float* cand_feat = (const float*)d_in[4];   // P x 32
  const float* g1_Wl   = (const float*)d_in[5];     // 128 x 128
  const float* g1_bl   = (const float*)d_in[6];
  const float* g1_Wr   = (const float*)d_in[7];
  const float* g1_br   = (const float*)d_in[8];
  const float* g1_We   = (const float*)d_in[9];     // 16 x 128
  const float* g1_att  = (const float*)d_in[10];    // 2 x 64
  const float* g1_bias = (const float*)d_in[11];    // 128
  const float* g2_Wl   = (const float*)d_in[12];    // 128 x 64
  const float* g2_bl   = (const float*)d_in[13];
  const float* g2_Wr   = (const float*)d_in[14];
  const float* g2_br   = (const float*)d_in[15];
  const float* g2_We   = (const float*)d_in[16];    // 16 x 64
  const float* g2_att  = (const float*)d_in[17];    // 64
  const float* g2_bias = (const float*)d_in[18];    // 64
  const float* em_W1   = (const float*)d_in[19];    // 32 x 64
  const float* em_b1   = (const float*)d_in[20];
  const float* em_W2   = (const float*)d_in[21];    // 64 x 64
  const float* em_b2   = (const float*)d_in[22];
  const float* sc_W1   = (const float*)d_in[23];    // 256 x 64
  const float* sc_b1   = (const float*)d_in[24];
  const float* sc_W2   = (const float*)d_in[25];    // 64 x 1
  const float* sc_b2   = (const float*)d_in[26];    // 1
  float* out = (float*)d_out;

  float* w = (float*)d_ws;
  const int* src = edge_idx;            // edge_index[0]
  const int* dst = edge_idx + NEDGES;   // edge_index[1]

  const int B = 256;
  const long nodeStripThreads = (long)(NNODES / 16) * 32;
  const long pairStripThreads = (long)(NPAIRS / 16) * 32;

  // ---- Stage 0: self-loop attrs -----------------------------------------
  hipMemsetAsync(w + F_LOOP, 0, (850000) * sizeof(float), stream); // loop+cnt
  deg_scatter<<<cdiv((long)NEDGES * EDGE_IN, B), B, 0, stream>>>(dst, edge_attr,
                                                                w + F_CNT, w + F_LOOP);
  loop_div<<<cdiv((long)NNODES * EDGE_IN, B), B, 0, stream>>>(w + F_LOOP, w + F_CNT);

  // ---- Stage 1: GAT layer 1 (heads=2, hd=128) ---------------------------
  wmma_gemm_rows<8><<<cdiv(nodeStripThreads, B), B, 0, stream>>>(
      x, g1_Wl, g1_bl, w + F_XL1, NNODES, 128, 0);
  wmma_gemm_rows<8><<<cdiv(nodeStripThreads, B), B, 0, stream>>>(
      x, g1_Wr, g1_br, w + F_XR1, NNODES, 128, 0);
  fill_i32<<<cdiv(NNODES * 2, B), B, 0, stream>>>((int*)(w + F_LM1), NNODES * 2, INT_MIN);
  hipMemsetAsync(w + F_DN1, 0, NNODES * 2 * sizeof(float), stream);
  hipMemsetAsync(w + F_H1, 0, (size_t)NNODES * 128 * sizeof(float), stream);

  gat_logits<2, 128><<<cdiv((long)E2 * 32, B), B, 0, stream>>>(
      src, dst, edge_attr, w + F_LOOP, w + F_XL1, w + F_XR1,
      g1_We, g1_att, w + F_LG1, (int*)(w + F_LM1));
  gat_expsum<2><<<cdiv((long)E2 * 2, B), B, 0, stream>>>(
      dst, w + F_LG1, (int*)(w + F_LM1), w + F_DN1);
  gat_agg<2, 128><<<cdiv((long)E2 * 128, B), B, 0, stream>>>(
      src, dst, w + F_LG1, w + F_DN1, w + F_XL1, w + F_H1);
  bias_relu<<<cdiv((long)NNODES * 128, B), B, 0, stream>>>(w + F_H1, g1_bias,
                                                           NNODES * 128, 128);

  // ---- Stage 2: GAT layer 2 (heads=1, hd=64) ----------------------------
  wmma_gemm_rows<4><<<cdiv(nodeStripThreads, B), B, 0, stream>>>(
      w + F_H1, g2_Wl, g2_bl, w + F_XL2, NNODES, 128, 0);
  wmma_gemm_rows<4><<<cdiv(nodeStripThreads, B), B, 0, stream>>>(
      w + F_H1, g2_Wr, g2_br, w + F_XR2, NNODES, 128, 0);
  fill_i32<<<cdiv(NNODES, B), B, 0, stream>>>((int*)(w + F_LM2), NNODES, INT_MIN);
  hipMemsetAsync(w + F_DN2, 0, NNODES * sizeof(float), stream);
  hipMemsetAsync(w + F_H2, 0, (size_t)NNODES * 64 * sizeof(float), stream);

  gat_logits<1, 64><<<cdiv((long)E2 * 32, B), B, 0, stream>>>(
      src, dst, edge_attr, w + F_LOOP, w + F_XL2, w + F_XR2,
      g2_We, g2_att, w + F_LG2, (int*)(w + F_LM2));
  gat_expsum<1><<<cdiv((long)E2, B), B, 0, stream>>>(
      dst, w + F_LG2, (int*)(w + F_LM2), w + F_DN2);
  gat_agg<1, 64><<<cdiv((long)E2 * 64, B), B, 0, stream>>>(
      src, dst, w + F_LG2, w + F_DN2, w + F_XL2, w + F_H2);
  bias_relu<<<cdiv((long)NNODES * 64, B), B, 0, stream>>>(w + F_H2, g2_bias,
                                                          NNODES * 64, 64);

  // ---- Stage 3: candidate scoring ---------------------------------------
  // hc1 = relu(cand_feat @ em.W1 + b1)
  wmma_gemm_rows<4><<<cdiv(pairStripThreads, B), B, 0, stream>>>(
      cand_feat, em_W1, em_b1, w + F_HC1, NPAIRS, 32, 1);
  // hc = relu(hc1 @ em.W2 + b2)
  wmma_gemm_rows<4><<<cdiv(pairStripThreads, B), B, 0, stream>>>(
      w + F_HC1, em_W2, em_b2, w + F_HC, NPAIRS, 64, 1);
  // s_hid = relu([hu|hv||hu-hv||hc] @ sc.W1 + b1), z synthesized in-flight
  score_hidden<<<cdiv(pairStripThreads, B), B, 0, stream>>>(
      pairs, w + F_H2, w + F_HC, sc_W1, sc_b1, w + F_SHID);
  score_final<<<cdiv((long)NPAIRS, B), B, 0, stream>>>(w + F_SHID, sc_W2, sc_b2, out);
}